// MolecularGNN_61297773249031
// MI455X (gfx1250) — compile-verified
//
#include <hip/hip_runtime.h>
#include <math.h>

typedef __attribute__((ext_vector_type(2))) float v2f;
typedef __attribute__((ext_vector_type(8))) float v8f;

__device__ __forceinline__ float leaky02(float x) { return x > 0.f ? x : 0.2f * x; }

// Order-preserving encoding of float into uint for atomicMax.
__device__ __forceinline__ unsigned encf(float f) {
    unsigned u = __float_as_uint(f);
    return (u & 0x80000000u) ? ~u : (u | 0x80000000u);
}
__device__ __forceinline__ float decf(unsigned e) {
    unsigned u = (e & 0x80000000u) ? (e & 0x7FFFFFFFu) : ~e;
    return __uint_as_float(u);
}

// ---------------------------------------------------------------------------
// C[N,128] = A[N,K] @ B[K,128] (+ bias), f32 WMMA 16x16x4. One wave computes a
// 16-row strip x 128 cols (8 v8f accumulators). B staged in LDS pre-swizzled:
// Bs[((k>>2)*128 + col)*4 + j] = B[k+j][col]  (j = 0..3)
// so each lane's B fragment is a single conflict-free ds_load_b64.
// ---------------------------------------------------------------------------
__global__ __launch_bounds__(256) void gemm_wmma(const float* __restrict__ A,
                                                 const float* __restrict__ B,
                                                 const float* __restrict__ bias,
                                                 float* __restrict__ C, int N, int K) {
    extern __shared__ float Bs[];  // K*128 floats
    const int tid = threadIdx.x;
    const int kq = K >> 2;
    for (int d = tid; d < kq * 128; d += 256) {
        const int p4 = d >> 7, col = d & 127;
        const float* sp = B + (p4 * 4) * 128 + col;
        ((float4*)Bs)[d] = make_float4(sp[0], sp[128], sp[256], sp[384]);
    }
    __syncthreads();

    const int wave = tid >> 5;
    const int lane = tid & 31;
    const int l16  = lane & 15;
    const int hi   = lane >> 4;
    const long row0 = (long)(blockIdx.x * 8 + wave) * 16;
    if (row0 >= N) return;

    v8f acc[8] = {};
    long arow = row0 + l16;
    if (arow >= N) arow = N - 1;  // clamp; OOB rows masked at store
    const float* Ap = A + arow * (long)K;
    const float* Bq = Bs + l16 * 4 + hi * 2;  // lane's column/half base

    for (int k0 = 0; k0 < K; k0 += 4) {
        // A frag (16x4): lanes 0-15 hold K=k0,k0+1; lanes 16-31 hold K=k0+2,k0+3
        v2f a;
        const float* ap = Ap + k0 + 2 * hi;
        a.x = ap[0];
        a.y = ap[1];
        const float* bp = Bq + (k0 >> 2) * 512;
#pragma unroll
        for (int t = 0; t < 8; ++t) {
            // B frag (4x16): V0 holds rows k0 / k0+2, V1 holds rows k0+1 / k0+3
            const v2f b = *(const v2f*)(bp + t * 64);  // single ds_load_b64
            acc[t] = __builtin_amdgcn_wmma_f32_16x16x4_f32(false, a, false, b,
                                                           (short)0, acc[t], false, false);
        }
    }
#pragma unroll
    for (int t = 0; t < 8; ++t) {
        const int col = t * 16 + l16;
        const float bb = bias ? bias[col] : 0.f;
#pragma unroll
        for (int r = 0; r < 8; ++r) {
            long row = row0 + r + 8 * hi;  // C/D layout: VGPR r -> rows r / r+8
            if (row < N) C[row * 128 + col] = acc[t][r] + bb;
        }
    }
}

// ---------------------------------------------------------------------------
// Fold per-layer weights: Vs/Vd[k,h]=sum_d W_lin[k,h*32+d]*att_{s,d}[h,d],
// Ve likewise from W_edge; M[16,4]=W_bond@Ve; c[4]=b_bond@Ve.
// ---------------------------------------------------------------------------
__global__ __launch_bounds__(128) void prep_layer(const float* __restrict__ Wl,
                                                  const float* __restrict__ We,
                                                  const float* __restrict__ as_,
                                                  const float* __restrict__ ad_,
                                                  const float* __restrict__ ae_,
                                                  const float* __restrict__ Wbond,
                                                  const float* __restrict__ bbond,
                                                  float* __restrict__ Vs, float* __restrict__ Vd,
                                                  float* __restrict__ Mmat,
                                                  float* __restrict__ cvec) {
    __shared__ float VeSh[128 * 4];
    __shared__ float asS[128], adS[128], aeS[128];
    const int t = threadIdx.x;  // 128 threads == rows of W
    asS[t] = as_[t];
    adS[t] = ad_[t];
    aeS[t] = ae_[t];
    __syncthreads();

    float vs[4] = {0, 0, 0, 0}, vd[4] = {0, 0, 0, 0}, ve[4] = {0, 0, 0, 0};
    for (int h = 0; h < 4; ++h)
        for (int d = 0; d < 32; ++d) {
            const int c = h * 32 + d;
            const float wl = Wl[t * 128 + c];
            const float we = We[t * 128 + c];
            vs[h] += wl * asS[c];
            vd[h] += wl * adS[c];
            ve[h] += we * aeS[c];
        }
    for (int h = 0; h < 4; ++h) {
        Vs[t * 4 + h] = vs[h];
        Vd[t * 4 + h] = vd[h];
        VeSh[t * 4 + h] = ve[h];
    }
    __syncthreads();
    if (t < 16) {
        float m[4] = {0, 0, 0, 0};
        for (int k = 0; k < 128; ++k) {
            const float w = Wbond[t * 128 + k];
            for (int h = 0; h < 4; ++h) m[h] += w * VeSh[k * 4 + h];
        }
        for (int h = 0; h < 4; ++h) Mmat[t * 4 + h] = m[h];
    } else if (t < 20) {
        const int h = t - 16;
        float cc = 0.f;
        for (int k = 0; k < 128; ++k) cc += bbond[k] * VeSh[k * 4 + h];
        cvec[h] = cc;
    }
}

// ---------------------------------------------------------------------------
// asrc/adst[n,h] = h[n,:] . Vs/Vd[:,h]. 32 nodes per block staged in LDS.
// ---------------------------------------------------------------------------
__global__ __launch_bounds__(128) void node_alpha(const float* __restrict__ hmat,
                                                  const float* __restrict__ Vs,
                                                  const float* __restrict__ Vd,
                                                  float* __restrict__ asrc,
                                                  float* __restrict__ adst, int N) {
    __shared__ float hs[32 * 132];
    __shared__ float VsS[512], VdS[512];
    const int t = threadIdx.x;
    for (int i = t; i < 512; i += 128) {
        VsS[i] = Vs[i];
        VdS[i] = Vd[i];
    }
    const long n0 = (long)blockIdx.x * 32;
    for (int i = t; i < 32 * 32; i += 128) {
        const int r = i >> 5, c4 = (i & 31) * 4;
        const long n = n0 + r;
        float4 v = make_float4(0.f, 0.f, 0.f, 0.f);
        if (n < N) v = *(const float4*)(hmat + n * 128 + c4);
        *(float4*)(hs + r * 132 + c4) = v;
    }
    __syncthreads();
    const int r = t >> 2, hh = t & 3;
    const long n = n0 + r;
    if (n >= N) return;
    float s = 0.f, d = 0.f;
    const float* hp = hs + r * 132;
    for (int k = 0; k < 128; ++k) {
        const float hv = hp[k];
        s += hv * VsS[k * 4 + hh];
        d += hv * VdS[k * 4 + hh];
    }
    asrc[n * 4 + hh] = s;
    adst[n * 4 + hh] = d;
}

__global__ void compute_deg(const int* __restrict__ dst, float* __restrict__ deg, int E) {
    const int e = blockIdx.x * blockDim.x + threadIdx.x;
    if (e < E) unsafeAtomicAdd(&deg[dst[e]], 1.0f);
}

// ae[e,h] = edge_attr[e,:] @ M + c ; accumulate into loopacc[dst]
__global__ __launch_bounds__(256) void edge_alpha_e(const float* __restrict__ eattr,
                                                    const int* __restrict__ dst,
                                                    const float* __restrict__ Mmat,
                                                    const float* __restrict__ cvec,
                                                    float* __restrict__ ae,
                                                    float* __restrict__ loopacc, int E) {
    __shared__ float Ms[64];
    __shared__ float cs[4];
    const int t = threadIdx.x;
    if (t < 64) Ms[t] = Mmat[t];
    if (t < 4) cs[t] = cvec[t];
    __syncthreads();
    const int e = blockIdx.x * 256 + t;
    if (e >= E) return;
    const float* ea = eattr + (long)e * 16;
    float a0 = cs[0], a1 = cs[1], a2 = cs[2], a3 = cs[3];
#pragma unroll
    for (int k = 0; k < 16; ++k) {
        const float v = ea[k];
        a0 += v * Ms[k * 4 + 0];
        a1 += v * Ms[k * 4 + 1];
        a2 += v * Ms[k * 4 + 2];
        a3 += v * Ms[k * 4 + 3];
    }
    const long o = (long)e * 4;
    ae[o + 0] = a0; ae[o + 1] = a1; ae[o + 2] = a2; ae[o + 3] = a3;
    const long dn = (long)dst[e] * 4;
    unsafeAtomicAdd(&loopacc[dn + 0], a0);
    unsafeAtomicAdd(&loopacc[dn + 1], a1);
    unsafeAtomicAdd(&loopacc[dn + 2], a2);
    unsafeAtomicAdd(&loopacc[dn + 3], a3);
}

// Self-loop logits; seeds amax (self-loop always present -> no -inf init).
__global__ void node_self(const float* __restrict__ asrc, const float* __restrict__ adst,
                          const float* __restrict__ loopacc, const float* __restrict__ deg,
                          float* __restrict__ aself, unsigned* __restrict__ amax, int N) {
    const int i = blockIdx.x * blockDim.x + threadIdx.x;
    if (i >= N * 4) return;
    const int n = i >> 2;
    const float inv = 1.0f / fmaxf(deg[n], 1.0f);
    const float a = leaky02(asrc[i] + adst[i] + loopacc[i] * inv);
    aself[i] = a;
    amax[i] = encf(a);
}

__global__ void edge_logits(const int* __restrict__ src, const int* __restrict__ dst,
                            const float* __restrict__ asrc, const float* __restrict__ adst,
                            float* __restrict__ ae, unsigned* __restrict__ amax, int E) {
    const int i = blockIdx.x * blockDim.x + threadIdx.x;  // (e, h)
    if (i >= E * 4) return;
    const int e = i >> 2, hh = i & 3;
    const int s = src[e], d = dst[e];
    const float a = leaky02(asrc[s * 4 + hh] + adst[d * 4 + hh] + ae[i]);
    ae[i] = a;
    atomicMax(&amax[d * 4 + hh], encf(a));
}

__global__ void node_expself(const unsigned* __restrict__ amax, float* __restrict__ aself,
                             float* __restrict__ denom, int N) {
    const int i = blockIdx.x * blockDim.x + threadIdx.x;
    if (i >= N * 4) return;
    const float w = expf(aself[i] - decf(amax[i]));
    aself[i] = w;
    denom[i] = w;  // denominator seeded with self term
}

__global__ void edge_exp(const int* __restrict__ dst, const unsigned* __restrict__ amax,
                         float* __restrict__ ae, float* __restrict__ denom, int E) {
    const int i = blockIdx.x * blockDim.x + threadIdx.x;  // (e, h)
    if (i >= E * 4) return;
    const int e = i >> 2, hh = i & 3;
    const int d = dst[e];
    const float w = expf(ae[i] - decf(amax[d * 4 + hh]));
    ae[i] = w;
    unsafeAtomicAdd(&denom[d * 4 + hh], w);
}

// acc[n,:] = xh[n,:] * wself_norm + bias  (plain-store init before edge scatter)
__global__ void node_selfagg(const float* __restrict__ xh, const float* __restrict__ aself,
                             const float* __restrict__ denom, const float* __restrict__ bias,
                             float* __restrict__ acc, int N) {
    const long i = (long)blockIdx.x * blockDim.x + threadIdx.x;
    if (i >= (long)N * 128) return;
    const int ch = (int)(i & 127);
    const long n = i >> 7;
    const int hh = ch >> 5;
    const float s = aself[n * 4 + hh] / (denom[n * 4 + hh] + 1e-16f);
    acc[i] = xh[i] * s + bias[ch];
}

// One wave per edge: gather xh[src] (float4/lane), scale per head, scatter-add.
__global__ __launch_bounds__(256) void edge_scatter(const int* __restrict__ src,
                                                    const int* __restrict__ dst,
                                                    const float* __restrict__ xh,
                                                    const float* __restrict__ ae,
                                                    const float* __restrict__ denom,
                                                    float* __restrict__ acc, int E) {
    const int e = (blockIdx.x * 256 + threadIdx.x) >> 5;
    const int lane = threadIdx.x & 31;
    if (e >= E) return;
    const int s = src[e], d = dst[e];
    const int hh = lane >> 3;  // (lane*4)/32
    const float w = ae[(long)e * 4 + hh] / (denom[(long)d * 4 + hh] + 1e-16f);
    const float4 xs = *(const float4*)(xh + (long)s * 128 + lane * 4);
    float* ap = acc + (long)d * 128 + lane * 4;
    unsafeAtomicAdd(ap + 0, xs.x * w);
    unsafeAtomicAdd(ap + 1, xs.y * w);
    unsafeAtomicAdd(ap + 2, xs.z * w);
    unsafeAtomicAdd(ap + 3, xs.w * w);
}

__global__ __launch_bounds__(128) void bn_stats(const float* __restrict__ acc,
                                                float* __restrict__ gsum,
                                                float* __restrict__ gsq, int N, int rpb) {
    const int ch = threadIdx.x;
    long r0 = (long)blockIdx.x * rpb;
    long r1 = r0 + rpb;
    if (r1 > N) r1 = N;
    float s = 0.f, q = 0.f;
    for (long r = r0; r < r1; ++r) {
        const float v = acc[r * 128 + ch];
        s += v;
        q += v * v;
    }
    unsafeAtomicAdd(&gsum[ch], s);
    unsafeAtomicAdd(&gsq[ch], q);
}

// h = relu(bn(acc)) + h   (fused apply + residual, in place on h)
__global__ void bn_apply(const float* __restrict__ acc, const float* __restrict__ gsum,
                         const float* __restrict__ gsq, const float* __restrict__ gamma,
                         const float* __restrict__ beta, float* __restrict__ h, int N) {
    const long i = (long)blockIdx.x * blockDim.x + threadIdx.x;
    if (i >= (long)N * 128) return;
    const int ch = (int)(i & 127);
    const float invN = 1.0f / (float)N;
    const float mu = gsum[ch] * invN;
    const float var = gsq[ch] * invN - mu * mu;
    float y = gamma[ch] * (acc[i] - mu) * rsqrtf(var + 1e-5f) + beta[ch];
    y = fmaxf(y, 0.f);
    h[i] = h[i] + y;
}

extern "C" void kernel_launch(void* const* d_in, const int* in_sizes, int n_in,
                              void* d_out, int out_size, void* d_ws, size_t ws_size,
                              hipStream_t stream) {
    (void)n_in; (void)out_size; (void)ws_size;
    const float* x       = (const float*)d_in[0];
    const int*   eidx    = (const int*)d_in[1];
    const float* eattr   = (const float*)d_in[2];
    const float* W_atom  = (const float*)d_in[3];
    const float* b_atom  = (const float*)d_in[4];
    const float* W_bond  = (const float*)d_in[5];
    const float* b_bond  = (const float*)d_in[6];
    const float* W_lin   = (const float*)d_in[7];
    const float* W_edge  = (const float*)d_in[8];
    const float* att_src = (const float*)d_in[9];
    const float* att_dst = (const float*)d_in[10];
    const float* att_edge= (const float*)d_in[11];
    const float* bias_cv = (const float*)d_in[12];
    const float* bn_g    = (const float*)d_in[13];
    const float* bn_b    = (const float*)d_in[14];
    const float* W_out   = (const float*)d_in[15];
    const float* b_out   = (const float*)d_in[16];

    const int N = in_sizes[0] / 64;
    const int E = in_sizes[2] / 16;
    const int* srcI = eidx;
    const int* dstI = eidx + E;

    char* p = (char*)d_ws;
    auto carve = [&](size_t bytes) -> char* {
        char* r = p;
        p += (bytes + 255) & ~(size_t)255;
        return r;
    };
    float*    h       = (float*)carve((size_t)N * 128 * 4);
    float*    xh      = (float*)carve((size_t)N * 128 * 4);
    float*    acc     = (float*)carve((size_t)N * 128 * 4);
    float*    ae      = (float*)carve((size_t)E * 4 * 4);
    float*    asrc    = (float*)carve((size_t)N * 4 * 4);
    float*    adst    = (float*)carve((size_t)N * 4 * 4);
    float*    aself   = (float*)carve((size_t)N * 4 * 4);
    unsigned* amax    = (unsigned*)carve((size_t)N * 4 * 4);
    float*    denom   = (float*)carve((size_t)N * 4 * 4);
    float*    loopacc = (float*)carve((size_t)N * 4 * 4);
    float*    deg     = (float*)carve((size_t)N * 4);
    float*    Vs      = (float*)carve(512 * 4);
    float*    Vd      = (float*)carve(512 * 4);
    float*    Mm      = (float*)carve(64 * 4);
    float*    cv      = (float*)carve(4 * 4);
    float*    gsum    = (float*)carve(128 * 4);
    float*    gsq     = (float*)carve(128 * 4);

    const int gemmBlocks = (N + 127) / 128;

    // h = x @ W_atom + b_atom   (K = 64)
    gemm_wmma<<<gemmBlocks, 256, 64 * 128 * 4, stream>>>(x, W_atom, b_atom, h, N, 64);

    hipMemsetAsync(deg, 0, (size_t)N * 4, stream);
    compute_deg<<<(E + 255) / 256, 256, 0, stream>>>(dstI, deg, E);

    for (int l = 0; l < 3; ++l) {
        const float* Wl = W_lin + (size_t)l * 128 * 128;
        const float* We = W_edge + (size_t)l * 128 * 128;
        prep_layer<<<1, 128, 0, stream>>>(Wl, We, att_src + l * 128, att_dst + l * 128,
                                          att_edge + l * 128, W_bond, b_bond, Vs, Vd, Mm, cv);
        // xh = h @ W_lin[l]  (K = 128)
        gemm_wmma<<<gemmBlocks, 256, 128 * 128 * 4, stream>>>(h, Wl, nullptr, xh, N, 128);
        node_alpha<<<(N + 31) / 32, 128, 0, stream>>>(h, Vs, Vd, asrc, adst, N);
        hipMemsetAsync(loopacc, 0, (size_t)N * 4 * 4, stream);
        edge_alpha_e<<<(E + 255) / 256, 256, 0, stream>>>(eattr, dstI, Mm, cv, ae, loopacc, E);
        node_self<<<(N * 4 + 255) / 256, 256, 0, stream>>>(asrc, adst, loopacc, deg, aself, amax, N);
        edge_logits<<<(E * 4 + 255) / 256, 256, 0, stream>>>(srcI, dstI, asrc, adst, ae, amax, E);
        node_expself<<<(N * 4 + 255) / 256, 256, 0, stream>>>(amax, aself, denom, N);
        edge_exp<<<(E * 4 + 255) / 256, 256, 0, stream>>>(dstI, amax, ae, denom, E);
        node_selfagg<<<(int)(((long)N * 128 + 255) / 256), 256, 0, stream>>>(
            xh, aself, denom, bias_cv + l * 128, acc, N);
        edge_scatter<<<(E + 7) / 8, 256, 0, stream>>>(srcI, dstI, xh, ae, denom, acc, E);
        hipMemsetAsync(gsum, 0, 128 * 4, stream);
        hipMemsetAsync(gsq, 0, 128 * 4, stream);
        const int rpb = 256;
        bn_stats<<<(N + rpb - 1) / rpb, 128, 0, stream>>>(acc, gsum, gsq, N, rpb);
        bn_apply<<<(int)(((long)N * 128 + 255) / 256), 256, 0, stream>>>(
            acc, gsum, gsq, bn_g + l * 128, bn_b + l * 128, h, N);
    }

    // out = h @ W_out + b_out
    gemm_wmma<<<gemmBlocks, 256, 128 * 128 * 4, stream>>>(h, W_out, b_out, (float*)d_out, N, 128);
}